// SpatialEncoding_76433237999805
// MI455X (gfx1250) — compile-verified
//
#include <hip/hip_runtime.h>
#include <stdint.h>

// Spatial encoding scatter for MI455X (gfx1250, wave32).
// Stage 1: zero-fill the NxN f32 output with b128 stores (RT policy -> stays in L2).
// Stage 2: scatter b[clip(len,1,5)-1] into out[src*N+dst] using the CDNA5 async
//          data mover: GLOBAL_STORE_ASYNC_FROM_LDS_B32 fuses the LDS table gather
//          with the global scatter (data never passes through VGPRs), tracked by
//          ASYNCcnt and fenced with s_wait_asynccnt.

typedef int   v4i __attribute__((ext_vector_type(4)));
typedef float v4f __attribute__((ext_vector_type(4)));

#define MAXPD 5

__global__ void __launch_bounds__(256)
spatial_fill_zero(float* __restrict__ out, unsigned n4, unsigned total) {
  unsigned gid    = blockIdx.x * blockDim.x + threadIdx.x;
  unsigned stride = gridDim.x * blockDim.x;
  v4f z = {0.0f, 0.0f, 0.0f, 0.0f};
  v4f* __restrict__ o4 = (v4f*)out;
  for (unsigned i = gid; i < n4; i += stride) {
    o4[i] = z;                       // global_store_b128
  }
  if (gid == 0) {                    // scalar tail (none for N=8192)
    for (unsigned i = (n4 << 2); i < total; ++i) out[i] = 0.0f;
  }
}

__device__ __forceinline__ void scatter_one(int s, int d, int L,
                                            unsigned lbase, unsigned N,
                                            float* __restrict__ out) {
  int idx = L - 1;                         // clip(len,1,MAXPD)-1
  idx = idx < 0 ? 0 : idx;
  idx = idx > (MAXPD - 1) ? (MAXPD - 1) : idx;
  unsigned goff = ((unsigned)s * N + (unsigned)d) * 4u;   // byte offset, < 2^31
  unsigned loff = lbase + (unsigned)idx * 4u;             // LDS byte address
  // Copies the float at LDS address loff to byte offset goff in out via the async data mover (no VGPR data).
  asm volatile("global_store_async_from_lds_b32 %0, %1, %2"
               :: "v"(goff), "v"(loff), "s"(out)
               : "memory");
}

__global__ void __launch_bounds__(256)
spatial_scatter_async(const int* __restrict__ src,
                      const int* __restrict__ dst,
                      const int* __restrict__ plen,
                      const float* __restrict__ bvec,
                      float* __restrict__ out,
                      unsigned E, unsigned N) {
  __shared__ float lb[MAXPD];
  if (threadIdx.x < MAXPD) lb[threadIdx.x] = bvec[threadIdx.x];
  __syncthreads();

  // AS3 -> flat cast: low 32 bits of the flat shared address are the LDS byte offset.
  unsigned lbase = (unsigned)(uintptr_t)&lb[0];

  unsigned gid = blockIdx.x * blockDim.x + threadIdx.x;
  unsigned nv  = E >> 2;                   // int4 groups
  if (gid < nv) {
    // Streamed once -> nontemporal so edge data does not evict the output from L2.
    v4i s4 = __builtin_nontemporal_load((const v4i*)src  + gid);
    v4i d4 = __builtin_nontemporal_load((const v4i*)dst  + gid);
    v4i l4 = __builtin_nontemporal_load((const v4i*)plen + gid);
#pragma unroll
    for (int j = 0; j < 4; ++j) {
      scatter_one(s4[j], d4[j], l4[j], lbase, N, out);
    }
  }

  unsigned tail = E & 3u;
  if (blockIdx.x == 0 && threadIdx.x < tail) {
    unsigned e = (nv << 2) + threadIdx.x;
    scatter_one(src[e], dst[e], plen[e], lbase, N, out);
  }

  // Drain the async data mover before the wave retires.
  asm volatile("s_wait_asynccnt 0" ::: "memory");
}

extern "C" void kernel_launch(void* const* d_in, const int* in_sizes, int n_in,
                              void* d_out, int out_size, void* d_ws, size_t ws_size,
                              hipStream_t stream) {
  // setup_inputs order: x[N,128], src[E], dst[E], path_len[E], b[5]
  const int*   src  = (const int*)d_in[1];
  const int*   dst  = (const int*)d_in[2];
  const int*   plen = (const int*)d_in[3];
  const float* bvec = (const float*)d_in[4];
  float*       out  = (float*)d_out;

  const unsigned E = (unsigned)in_sizes[1];
  const unsigned N = (unsigned)(in_sizes[0] / 128);

  // ---- Stage 1: zero-fill (256 MB @ HBM store bandwidth) ----
  const unsigned total = (unsigned)out_size;
  const unsigned n4    = total >> 2;
  unsigned fb = (n4 + 255u) / 256u;
  if (fb > 8192u) fb = 8192u;              // grid-stride beyond this
  if (fb == 0u)   fb = 1u;
  spatial_fill_zero<<<fb, 256, 0, stream>>>(out, n4, total);

  // ---- Stage 2: async-scatter (stream-ordered after the fill) ----
  const unsigned nv = E >> 2;
  unsigned sb = (nv + 255u) / 256u;
  if (sb == 0u) sb = 1u;
  spatial_scatter_async<<<sb, 256, 0, stream>>>(src, dst, plen, bvec, out, E, N);
}